// EMAVectorQuantizer_42262478193133
// MI455X (gfx1250) — compile-verified
//
#include <hip/hip_runtime.h>
#include <hip/hip_bf16.h>
#include <math.h>

// ---------------- problem constants ----------------
#define NVEC   32768      // B*H*W
#define DDIM   64         // C
#define KCB    8192       // codebook entries
#define HW     1024       // H*W
#define PLANE  65536      // C*H*W
#define NB     32         // batch
#define TOTELEM 2097152   // B*C*H*W

#define DECAY  0.99f
#define EPSV   1e-5f

// search-kernel tiling
#define MT         128    // vectors per block (8 waves x 16 rows)
#define KC         128    // codes per LDS chunk
#define LDSS       68     // padded row stride (floats) -> conflict-free

// output layout (floats, concatenated in reference return order)
#define Q_OFF    0
#define IDX_OFF  2097152
#define LOSS_OFF 2129920
#define PERP_OFF 2129921
#define ACT_OFF  2129922
#define NCB_OFF  2129923
#define NCS_OFF  2654211
#define NEA_OFF  2662403

// workspace layout (floats)
#define WS_C2     0                      // [KCB]    0.5*|c|^2
#define WS_CNT    8192                   // [KCB]    counts
#define WS_LOSS   16384                  // scalar
#define WS_NTOT   16385                  // scalar
#define WS_PERP   16386                  // scalar
#define WS_ACT    16387                  // scalar
#define WS_SUMS   16388                  // [KCB*DDIM]
#define WS_ZERO_BEG 8192
#define WS_ZERO_END (16388 + KCB*DDIM)   // 540676

typedef float v2f __attribute__((ext_vector_type(2)));
typedef float v8f __attribute__((ext_vector_type(8)));

// ---------------- 1) zero scratch ----------------
__global__ void vq_zero_kernel(float* ws) {
    int gid = blockIdx.x * blockDim.x + threadIdx.x + WS_ZERO_BEG;
    if (gid < WS_ZERO_END) ws[gid] = 0.0f;
}

// ---------------- 2) 0.5*|codebook row|^2 ----------------
__global__ void vq_c2_kernel(const float* __restrict__ cb, float* __restrict__ c2h) {
    int k = blockIdx.x * blockDim.x + threadIdx.x;
    if (k < KCB) {
        const float* row = cb + (size_t)k * DDIM;
        float s = 0.0f;
        #pragma unroll
        for (int c = 0; c < DDIM; ++c) s = fmaf(row[c], row[c], s);
        c2h[k] = 0.5f * s;
    }
}

// ---------------- 3) WMMA nearest-neighbor search ----------------
// score(n,k) = 0.5*|c_k|^2 - v_n . c_k   (same argmin as squared L2)
__global__ __launch_bounds__(256)
void vq_search_kernel(const float* __restrict__ z, const float* __restrict__ cb,
                      const float* __restrict__ c2h, float* __restrict__ out_idx_f,
                      float* __restrict__ counts)
{
    __shared__ float lds_a[MT * LDSS];
    __shared__ float lds_b[KC * LDSS];
    __shared__ float lds_c2[KC];

    const int tid  = threadIdx.x;
    const int lane = tid & 31;
    const int wave = tid >> 5;      // 0..7
    const int half = lane >> 4;     // 0/1: K-slice half (A/B), row half (C/D)
    const int l16  = lane & 15;

    const int n0  = blockIdx.x * MT;       // tile-aligned inside an HW plane
    const int b   = n0 / HW;
    const int hw0 = n0 % HW;
    const float* zb = z + (size_t)b * PLANE + hw0;

    // stage A tile [MT x DDIM] from strided z layout (coalesced per c-plane)
    #pragma unroll 4
    for (int it = 0; it < 32; ++it) {
        int c = it * 2 + (tid >> 7);
        int i = tid & 127;
        lds_a[i * LDSS + c] = zb[c * HW + i];
    }
    __syncthreads();

    // A fragments: 16 k-slices of 4; lanes 0-15 hold K={0,1}, 16-31 hold K={2,3}
    v2f afrag[16];
    {
        const float* ar = &lds_a[(wave * 16 + l16) * LDSS];
        const int o0 = half * 2;
        #pragma unroll
        for (int f = 0; f < 16; ++f) {
            afrag[f].x = ar[f * 4 + o0];
            afrag[f].y = ar[f * 4 + o0 + 1];
        }
    }

    float best[8];
    int   bestk[8];
    #pragma unroll
    for (int r = 0; r < 8; ++r) { best[r] = 3.4e38f; bestk[r] = 0; }

    for (int kbase = 0; kbase < KCB; kbase += KC) {
        __syncthreads();
        #pragma unroll 4
        for (int it = 0; it < 32; ++it) {
            int row = it * 4 + (tid >> 6);
            int c   = tid & 63;
            lds_b[row * LDSS + c] = cb[(size_t)(kbase + row) * DDIM + c];
        }
        if (tid < KC) lds_c2[tid] = c2h[kbase + tid];
        __syncthreads();

        #pragma unroll
        for (int sub = 0; sub < KC / 16; ++sub) {
            const float* br = &lds_b[(sub * 16 + l16) * LDSS];
            const int o0 = half * 2;
            v8f acc = {};
            #pragma unroll
            for (int f = 0; f < 16; ++f) {
                v2f bfrag;
                bfrag.x = br[f * 4 + o0];
                bfrag.y = br[f * 4 + o0 + 1];
                // D = A(16x4) * B(4x16) + C   -> v_wmma_f32_16x16x4_f32
                acc = __builtin_amdgcn_wmma_f32_16x16x4_f32(
                          false, afrag[f], false, bfrag, (short)0, acc, false, false);
            }
            const float cj   = lds_c2[sub * 16 + l16];   // column = l16
            const int   ktag = kbase + sub * 16;
            #pragma unroll
            for (int r = 0; r < 8; ++r) {
                float s = cj - acc[r];
                if (s < best[r]) { best[r] = s; bestk[r] = ktag; }
            }
        }
    }

    // per-row argmin across the 16 column-lanes of each half (first-index tiebreak)
    #pragma unroll
    for (int r = 0; r < 8; ++r) {
        float v = best[r];
        int   k = bestk[r] + l16;
        #pragma unroll
        for (int off = 8; off >= 1; off >>= 1) {
            float ov = __shfl_xor(v, off, 32);
            int   ok = __shfl_xor(k, off, 32);
            if (ov < v || (ov == v && ok < k)) { v = ov; k = ok; }
        }
        if (l16 == 0) {
            int n = n0 + wave * 16 + half * 8 + r;   // C/D: VGPR r -> row r (+8 for hi half)
            out_idx_f[n] = (float)k;
            atomicAdd(&counts[k], 1.0f);
        }
    }
}

// ---------------- 4) quantized output + loss + scatter sums ----------------
__global__ __launch_bounds__(256)
void vq_apply_kernel(const float* __restrict__ z, const float* __restrict__ cb,
                     const float* __restrict__ idx_f, float* __restrict__ out_q,
                     float* __restrict__ sums, float* __restrict__ loss_acc)
{
    int gid = blockIdx.x * blockDim.x + threadIdx.x;  // = b*PLANE + c*HW + hw
    int b   = gid >> 16;
    int rem = gid & 65535;
    int c   = rem >> 10;
    int hw  = rem & 1023;
    int n   = b * HW + hw;
    int k   = (int)idx_f[n];

    float zv = z[gid];
    float q  = cb[(size_t)k * DDIM + c];
    out_q[gid] = q;                                   // straight-through fwd value
    atomicAdd(&sums[(size_t)k * DDIM + c], zv);       // segment_sum of vectors

    float d  = q - zv;
    float sq = d * d;
    #pragma unroll
    for (int off = 16; off >= 1; off >>= 1) sq += __shfl_xor(sq, off, 32);
    if ((threadIdx.x & 31) == 0) atomicAdd(loss_acc, sq);
}

// ---------------- 5) per-code stats + new_embed_avg ----------------
__global__ void vq_stats_kernel(const float* __restrict__ cluster_size,
                                const float* __restrict__ embed_avg,
                                const float* __restrict__ counts,
                                const float* __restrict__ sums,
                                float* __restrict__ out_nea, float* __restrict__ out_ncs,
                                float* ntot, float* perp, float* act)
{
    int gid = blockIdx.x * blockDim.x + threadIdx.x;  // [0, KCB*DDIM)
    out_nea[gid] = DECAY * embed_avg[gid] + (1.0f - DECAY) * sums[gid];
    if (gid < KCB) {
        float cnt = counts[gid];
        float ncs = DECAY * cluster_size[gid] + (1.0f - DECAY) * cnt;
        out_ncs[gid] = ncs;
        atomicAdd(ntot, ncs);
        if (cnt > 0.0f) atomicAdd(act, 1.0f);
        float p = cnt / (float)NVEC;
        atomicAdd(perp, -p * logf(p + 1e-10f));
    }
}

// ---------------- 6) smoothed codebook + scalars ----------------
__global__ void vq_final_kernel(const float* __restrict__ out_ncs,
                                const float* __restrict__ out_nea,
                                float* __restrict__ out_ncb,
                                const float* ntot, const float* perp,
                                const float* act, const float* loss_acc,
                                float* __restrict__ out_scalars)
{
    int gid = blockIdx.x * blockDim.x + threadIdx.x;  // [0, KCB*DDIM)
    float n_total = *ntot;
    int k = gid >> 6;
    float sm = (out_ncs[k] + EPSV) / (n_total + (float)KCB * EPSV) * n_total;
    out_ncb[gid] = out_nea[gid] / sm;
    if (gid == 0) {
        out_scalars[0] = loss_acc[0] / (float)TOTELEM;  // vq_loss
        out_scalars[1] = expf(perp[0]);                 // perplexity
        out_scalars[2] = act[0];                        // active_codes
    }
}

extern "C" void kernel_launch(void* const* d_in, const int* in_sizes, int n_in,
                              void* d_out, int out_size, void* d_ws, size_t ws_size,
                              hipStream_t stream) {
    const float* z            = (const float*)d_in[0];
    const float* codebook     = (const float*)d_in[1];
    const float* cluster_size = (const float*)d_in[2];
    const float* embed_avg    = (const float*)d_in[3];
    float* out = (float*)d_out;
    float* ws  = (float*)d_ws;

    float* c2h      = ws + WS_C2;
    float* counts   = ws + WS_CNT;
    float* loss_acc = ws + WS_LOSS;
    float* ntot     = ws + WS_NTOT;
    float* perp     = ws + WS_PERP;
    float* act      = ws + WS_ACT;
    float* sums     = ws + WS_SUMS;

    // 1) zero scratch accumulators
    {
        int n = WS_ZERO_END - WS_ZERO_BEG;
        vq_zero_kernel<<<(n + 255) / 256, 256, 0, stream>>>(ws);
    }
    // 2) code norms
    vq_c2_kernel<<<KCB / 256, 256, 0, stream>>>(codebook, c2h);
    // 3) WMMA nearest-neighbor search -> indices (as float) + counts
    vq_search_kernel<<<NVEC / MT, 256, 0, stream>>>(z, codebook, c2h,
                                                    out + IDX_OFF, counts);
    // 4) quantized output, commitment loss, per-code vector sums
    vq_apply_kernel<<<TOTELEM / 256, 256, 0, stream>>>(z, codebook, out + IDX_OFF,
                                                       out + Q_OFF, sums, loss_acc);
    // 5) EMA stats
    vq_stats_kernel<<<(KCB * DDIM) / 256, 256, 0, stream>>>(cluster_size, embed_avg,
                                                            counts, sums,
                                                            out + NEA_OFF, out + NCS_OFF,
                                                            ntot, perp, act);
    // 6) smoothed codebook + scalar outputs
    vq_final_kernel<<<(KCB * DDIM) / 256, 256, 0, stream>>>(out + NCS_OFF, out + NEA_OFF,
                                                            out + NCB_OFF, ntot, perp, act,
                                                            loss_acc, out + LOSS_OFF);
}